// VectorQuantizer_17995912970291
// MI455X (gfx1250) — compile-verified
//
#include <hip/hip_runtime.h>

typedef __attribute__((ext_vector_type(2))) float v2f;
typedef __attribute__((ext_vector_type(4))) float v4f;
typedef __attribute__((ext_vector_type(8))) float v8f;

#define NTOK  65536
#define KCB   8192
#define DDIM  64
#define DECAY 0.9f
#define OMD   0.1f
#define EPSV  1e-5f

// ---------------- workspace layout (4-byte units) ----------------
#define WS_W2      0                         // [KCB] float : ||c_k||^2
#define WS_TOKEN   (WS_W2 + KCB)             // [NTOK] int
#define WS_ROWERR  (WS_TOKEN + NTOK)         // [NTOK] float
#define WS_COUNTS  (WS_ROWERR + NTOK)        // [KCB] int
#define WS_ESUM    (WS_COUNTS + KCB)         // [KCB*DDIM] float
#define WS_NTOTAL  (WS_ESUM + KCB*DDIM)      // [1] float

// ---------------- output layout ----------------
#define OUT_QERR 0
#define OUT_W    1
#define OUT_NCS  (1 + KCB*DDIM)
#define OUT_NEA  (1 + KCB*DDIM + KCB)

// ================= zero the atomic accumulators =================
__global__ void vq_zero(int* __restrict__ wsi) {
    int i = blockIdx.x * 256 + threadIdx.x;
    if (i < KCB + KCB * DDIM) wsi[WS_COUNTS + i] = 0;
}

// ================= ||c_k||^2 =================
__global__ void vq_w2(const float* __restrict__ w, float* __restrict__ w2) {
    int k = blockIdx.x * 256 + threadIdx.x;
    if (k < KCB) {
        const v4f* p = (const v4f*)(w + (size_t)k * DDIM);
        float s = 0.f;
#pragma unroll
        for (int i = 0; i < DDIM / 4; ++i) {
            v4f v = p[i];
            s += v.x * v.x + v.y * v.y + v.z * v.z + v.w * v.w;
        }
        w2[k] = s;
    }
}

// ================= fused distance + argmin (WMMA f32 16x16x4) =================
// one wave -> 16 rows of z; workgroup (8 waves) shares weight chunks via a
// double-buffered LDS staging area filled with async global->LDS copies.
#define CHUNK_CODES 128
#define LDS_STRIDE  68   // 68*16 = 1088 = 0 mod 64 -> conflict-free b64 reads
#define NCHUNK      (KCB / CHUNK_CODES)

// Issue 8 async B128 copies (this thread's share of one 128x64 f32 chunk).
__device__ __forceinline__ void issue_chunk_async(const float* __restrict__ weight,
                                                  int k0, float* buf, int tid)
{
    const char* gbase = (const char*)(weight + (size_t)k0 * DDIM);
#pragma unroll
    for (int r = 0; r < 8; ++r) {
        int q    = tid + r * 256;        // float4 index within the chunk
        int code = q >> 4;
        int off  = (q & 15) * 4;
        unsigned lds = (unsigned)(uintptr_t)&buf[code * LDS_STRIDE + off];
        unsigned long long ga = (unsigned long long)(uintptr_t)gbase
                              + (unsigned long long)q * 16ull;
        asm volatile("global_load_async_to_lds_b128 %0, %1, off"
                     :: "v"(lds), "v"(ga) : "memory");
    }
}

__global__ __launch_bounds__(256)
void vq_argmin(const float* __restrict__ z, const float* __restrict__ weight,
               const float* __restrict__ w2, int* __restrict__ token,
               float* __restrict__ row_err)
{
    __shared__ float lw[2][CHUNK_CODES * LDS_STRIDE];

    const int tid  = threadIdx.x;
    const int lane = tid & 31;
    const int wave = tid >> 5;
    const int half = lane >> 4;   // which 16-lane half of the wave32
    const int l16  = lane & 15;
    const int row0 = (blockIdx.x * 8 + wave) * 16;

    // prefetch chunk 0 into buffer 0 (8 async B128 per thread)
    issue_chunk_async(weight, 0, lw[0], tid);

    // A-matrix (16x4 f32) layout: lanes 0-15 = rows, VGPR0/1 = K{0,1}, upper half K{2,3}
    v2f a[16];
    float zsq = 0.f;
    {
        const float* zr = z + (size_t)(row0 + l16) * DDIM + 2 * half;
#pragma unroll
        for (int c = 0; c < 16; ++c) {
            float x0 = zr[4 * c + 0];
            float x1 = zr[4 * c + 1];
            a[c].x = x0; a[c].y = x1;
            zsq += x0 * x0 + x1 * x1;
        }
    }
    zsq += __shfl_xor(zsq, 16);   // full ||z_row||^2 for row row0+l16 on every lane

    float runmin[8];
    int   runidx[8];
#pragma unroll
    for (int j = 0; j < 8; ++j) { runmin[j] = __builtin_inff(); runidx[j] = 0; }

    for (int ci = 0; ci < NCHUNK; ++ci) {
        float* cur = lw[ci & 1];
        const bool has_next = (ci + 1) < NCHUNK;
        if (has_next) {
            // safe: the barrier at the end of iteration ci-1 guarantees no wave
            // is still reading lw[(ci+1)&1]
            issue_chunk_async(weight, (ci + 1) * CHUNK_CODES, lw[(ci + 1) & 1], tid);
            asm volatile("s_wait_asynccnt 0x8" ::: "memory");  // batch ci done (in-order)
        } else {
            asm volatile("s_wait_asynccnt 0x0" ::: "memory");
        }
        __syncthreads();   // all waves' copies of chunk ci are in LDS

#pragma unroll 1
        for (int kt = 0; kt < CHUNK_CODES / 16; ++kt) {
            // B-matrix (4x16 f32): lanes 0-15 = codes (N), VGPR0/1 = K{0,1}, upper half K{2,3}
            const float* wl = &cur[(kt * 16 + l16) * LDS_STRIDE + 2 * half];
            v8f acc0 = {}, acc1 = {};   // two independent chains -> half the RAW depth
#pragma unroll
            for (int c = 0; c < 16; c += 2) {
                v2f b0, b1;
                b0.x = wl[4 * c + 0]; b0.y = wl[4 * c + 1];
                b1.x = wl[4 * c + 4]; b1.y = wl[4 * c + 5];
                acc0 = __builtin_amdgcn_wmma_f32_16x16x4_f32(
                    false, a[c],     false, b0, (short)0, acc0, false, false);
                acc1 = __builtin_amdgcn_wmma_f32_16x16x4_f32(
                    false, a[c + 1], false, b1, (short)0, acc1, false, false);
            }
            int   code = ci * CHUNK_CODES + kt * 16 + l16;
            float w2v  = w2[code];
#pragma unroll
            for (int j = 0; j < 8; ++j) {
                float s = fmaf(-2.f, acc0[j] + acc1[j], w2v); // dist - ||z||^2
                if (s < runmin[j]) { runmin[j] = s; runidx[j] = code; }
            }
        }
        __syncthreads();   // all waves done reading cur before batch ci+2 overwrites it
    }

    // cross-lane argmin within each 16-lane half (first-index tie-break)
#pragma unroll
    for (int j = 0; j < 8; ++j) {
#pragma unroll
        for (int off = 8; off >= 1; off >>= 1) {
            float om = __shfl_xor(runmin[j], off);
            int   oi = __shfl_xor(runidx[j], off);
            if (om < runmin[j] || (om == runmin[j] && oi < runidx[j])) {
                runmin[j] = om; runidx[j] = oi;
            }
        }
    }

    // row r lives in slot j = r&7 of half r>>3; lane with l16==r in that half writes it
    if ((l16 >> 3) == half) {
        int j = l16 & 7;
        token[row0 + l16] = runidx[j];
        float e = zsq + runmin[j];            // = ||z - c_argmin||^2
        row_err[row0 + l16] = e > 0.f ? e : 0.f;
    }
}

// ================= segment sums via atomics (coalesced over d) =================
__global__ void vq_scatter(const float* __restrict__ z, const int* __restrict__ token,
                           int* __restrict__ counts, float* __restrict__ esum)
{
    int idx = blockIdx.x * 256 + threadIdx.x;   // over N*D
    int n = idx >> 6;
    int d = idx & 63;
    int t = token[n];
    if (d == 0) atomicAdd(&counts[t], 1);
    atomicAdd(&esum[t * DDIM + d], z[idx]);
}

// ================= quant_error = mean(row_err) (fixed-order tree) =================
__global__ void vq_qerr(const float* __restrict__ row_err, float* __restrict__ out)
{
    __shared__ float sm[256];
    int tid = threadIdx.x;
    float s = 0.f;
    for (int i = tid; i < NTOK; i += 256) s += row_err[i];
    sm[tid] = s;
    __syncthreads();
    for (int st = 128; st > 0; st >>= 1) {
        if (tid < st) sm[tid] += sm[tid + st];
        __syncthreads();
    }
    if (tid == 0) out[OUT_QERR] = sm[0] * (1.0f / NTOK);
}

// ================= new_cluster_size + its sum (single block, deterministic) =====
__global__ void vq_ncs(const float* __restrict__ cs, const int* __restrict__ counts,
                       float* __restrict__ out_ncs, float* __restrict__ n_total)
{
    __shared__ float sm[1024];
    int tid = threadIdx.x;
    float s = 0.f;
    for (int k = tid; k < KCB; k += 1024) {
        float v = cs[k] * DECAY + OMD * (float)counts[k];
        out_ncs[k] = v;
        s += v;
    }
    sm[tid] = s;
    __syncthreads();
    for (int st = 512; st > 0; st >>= 1) {
        if (tid < st) sm[tid] += sm[tid + st];
        __syncthreads();
    }
    if (tid == 0) *n_total = sm[0];
}

// ================= new_embed_avg + new_weight =================
__global__ void vq_final(const float* __restrict__ ea, const float* __restrict__ esum,
                         const float* __restrict__ ncs, const float* __restrict__ n_total,
                         float* __restrict__ out_w, float* __restrict__ out_nea)
{
    int idx = blockIdx.x * 256 + threadIdx.x;   // over K*D
    int k = idx >> 6;
    float nea = ea[idx] * DECAY + OMD * esum[idx];
    out_nea[idx] = nea;
    float n = *n_total;
    float smoothed = (ncs[k] + EPSV) / (n + (float)KCB * EPSV) * n;
    out_w[idx] = nea / smoothed;
}

extern "C" void kernel_launch(void* const* d_in, const int* in_sizes, int n_in,
                              void* d_out, int out_size, void* d_ws, size_t ws_size,
                              hipStream_t stream)
{
    (void)in_sizes; (void)n_in; (void)out_size; (void)ws_size;
    const float* z   = (const float*)d_in[0];
    const float* wgt = (const float*)d_in[1];
    const float* cs  = (const float*)d_in[2];
    const float* ea  = (const float*)d_in[3];
    float* out = (float*)d_out;
    float* wsf = (float*)d_ws;
    int*   wsi = (int*)d_ws;

    float* w2      = wsf + WS_W2;
    int*   token   = wsi + WS_TOKEN;
    float* row_err = wsf + WS_ROWERR;
    int*   counts  = wsi + WS_COUNTS;
    float* esum    = wsf + WS_ESUM;
    float* ntotal  = wsf + WS_NTOTAL;

    vq_zero   <<<(KCB + KCB * DDIM + 255) / 256, 256, 0, stream>>>(wsi);
    vq_w2     <<<KCB / 256, 256, 0, stream>>>(wgt, w2);
    vq_argmin <<<NTOK / (16 * 8), 256, 0, stream>>>(z, wgt, w2, token, row_err);
    vq_scatter<<<(NTOK * DDIM) / 256, 256, 0, stream>>>(z, token, counts, esum);
    vq_qerr   <<<1, 256, 0, stream>>>(row_err, out);
    vq_ncs    <<<1, 1024, 0, stream>>>(cs, counts, out + OUT_NCS, ntotal);
    vq_final  <<<(KCB * DDIM) / 256, 256, 0, stream>>>(ea, esum, out + OUT_NCS, ntotal,
                                                       out + OUT_W, out + OUT_NEA);
}